// MyHypergraphAtt_23089744183325
// MI455X (gfx1250) — compile-verified
//
#include <hip/hip_runtime.h>
#include <hip/hip_bf16.h>
#include <math.h>

// Problem geometry (fixed by the reference)
#define NN 2048   // nodes
#define NE 1024   // edges
#define DD 64     // feature dim
#define DO 64     // output dim
#define HH 128    // 2*DD hidden dim

typedef __attribute__((ext_vector_type(2))) float v2f;
typedef __attribute__((ext_vector_type(8))) float v8f;

// f32 WMMA: D(16x16,f32) = A(16x4,f32) * B(4x16,f32) + C
__device__ __forceinline__ v8f wmma_f32(v2f a, v2f b, v8f c) {
  return __builtin_amdgcn_wmma_f32_16x16x4_f32(
      /*neg_a=*/false, a, /*neg_b=*/false, b,
      /*c_mod=*/(short)0, c, /*reuse_a=*/false, /*reuse_b=*/false);
}

// ---------------- degrees ----------------
__global__ void k_deg_n(const float* __restrict__ H, float* __restrict__ dn) {
  int row  = blockIdx.x * (blockDim.x >> 5) + (threadIdx.x >> 5);
  int lane = threadIdx.x & 31;
  if (row >= NN) return;
  const float* hr = H + (size_t)row * NE;
  float s = 0.f;
  for (int e = lane; e < NE; e += 32) s += hr[e];
  #pragma unroll
  for (int off = 16; off > 0; off >>= 1) s += __shfl_down(s, off, 32);
  if (lane == 0) dn[row] = (s > 0.f) ? rsqrtf(s) : 0.f;
}

__global__ void k_deg_e(const float* __restrict__ H, float* __restrict__ de) {
  int e = blockIdx.x * blockDim.x + threadIdx.x;
  if (e >= NE) return;
  float s = 0.f;
  for (int n = 0; n < NN; ++n) s += H[(size_t)n * NE + e];
  de[e] = (s > 0.f) ? (1.f / s) : 0.f;
}

// ---------------- ef = H^T @ x  [NE, DD] ----------------
// one wave per 16x16 tile; K = NN; software-pipelined (double-buffered) loads
__global__ void k_edge_feat(const float* __restrict__ H, const float* __restrict__ x,
                            float* __restrict__ ef) {
  int wave = threadIdx.x >> 5, lane = threadIdx.x & 31;
  int tile = blockIdx.x * (blockDim.x >> 5) + wave;
  const int ntD = DD / 16;
  int mt = tile / ntD, nt = tile % ntD;
  int half = lane >> 4, lm = lane & 15;
  int e = mt * 16 + lm;                  // A row (edge)
  int d = nt * 16 + lm;                  // B col (dim)

  v8f c = {};
  v2f a, b;
  {
    int ka = half * 2;
    a.x = H[(size_t)ka * NE + e];
    a.y = H[(size_t)(ka + 1) * NE + e];
    b.x = x[(size_t)ka * DD + d];
    b.y = x[(size_t)(ka + 1) * DD + d];
  }
  for (int k0 = 0; k0 < NN - 4; k0 += 4) {
    int ka = k0 + 4 + half * 2;
    v2f an, bn;
    an.x = H[(size_t)ka * NE + e];
    an.y = H[(size_t)(ka + 1) * NE + e];
    bn.x = x[(size_t)ka * DD + d];
    bn.y = x[(size_t)(ka + 1) * DD + d];
    int pk = k0 + 32; pk = (pk > NN - 4) ? (NN - 4) : pk;   // branch-free clamp
    __builtin_prefetch(H + (size_t)(pk + half * 2) * NE + e, 0, 3);
    c = wmma_f32(a, b, c);
    a = an; b = bn;
  }
  c = wmma_f32(a, b, c);

  int mb = mt * 16 + half * 8;
  #pragma unroll
  for (int r = 0; r < 8; ++r)
    ef[(size_t)(mb + r) * DD + d] = c[r];
}

// ---------------- np = x @ W1[:, :D]^T ; ep = ef @ W1[:, D:]^T ----------------
__global__ void k_proj(const float* __restrict__ x, const float* __restrict__ ef,
                       const float* __restrict__ W1,
                       float* __restrict__ np, float* __restrict__ ep) {
  int wave = threadIdx.x >> 5, lane = threadIdx.x & 31;
  int tile = blockIdx.x * (blockDim.x >> 5) + wave;
  const int jt = HH / 16;                 // 8 output-col tiles
  const int npTiles = (NN / 16) * jt;     // 1024
  const float* A; float* C; int koff;
  if (tile < npTiles) { A = x;  C = np; koff = 0; }
  else { tile -= npTiles; A = ef; C = ep; koff = DD; }
  int mt = tile / jt, nt = tile % jt;
  int half = lane >> 4, lm = lane & 15;
  int m = mt * 16 + lm;
  int j = nt * 16 + lm;
  const float* Arow = A + (size_t)m * DD;
  const float* Brow = W1 + (size_t)j * HH + koff;
  v8f c = {};
  #pragma unroll
  for (int k0 = 0; k0 < DD; k0 += 4) {
    int ka = k0 + half * 2;
    v2f a = *(const v2f*)(Arow + ka);     // contiguous -> global_load_b64
    v2f b = *(const v2f*)(Brow + ka);
    c = wmma_f32(a, b, c);
  }
  int mb = mt * 16 + half * 8;
  #pragma unroll
  for (int r = 0; r < 8; ++r)
    C[(size_t)(mb + r) * HH + j] = c[r];
}

// ---------------- H_att = sigmoid(score) * H, only at H != 0 ----------------
__global__ void k_score(const float* __restrict__ H, const float* __restrict__ np,
                        const float* __restrict__ ep, const float* __restrict__ b1,
                        const float* __restrict__ W2, const float* __restrict__ b2,
                        float* __restrict__ Hatt) {
  __shared__ float s_np[HH], s_w2[HH], s_b1[HH];
  int n = blockIdx.x;
  for (int i = threadIdx.x; i < HH; i += blockDim.x) {
    s_np[i] = np[(size_t)n * HH + i];
    s_w2[i] = W2[i];
    s_b1[i] = b1[i];
  }
  __syncthreads();
  float bias2 = b2[0];
  for (int e = threadIdx.x; e < NE; e += blockDim.x) {
    float h = H[(size_t)n * NE + e];
    float out = 0.f;
    if (h != 0.f) {
      float s = bias2;
      const float4* epr = (const float4*)(ep + (size_t)e * HH);
      #pragma unroll 4
      for (int k4 = 0; k4 < HH / 4; ++k4) {
        float4 ev = epr[k4];              // global_load_b128
        int k = k4 * 4;
        s = fmaf(fmaxf(s_np[k + 0] + ev.x + s_b1[k + 0], 0.f), s_w2[k + 0], s);
        s = fmaf(fmaxf(s_np[k + 1] + ev.y + s_b1[k + 1], 0.f), s_w2[k + 1], s);
        s = fmaf(fmaxf(s_np[k + 2] + ev.z + s_b1[k + 2], 0.f), s_w2[k + 2], s);
        s = fmaf(fmaxf(s_np[k + 3] + ev.w + s_b1[k + 3], 0.f), s_w2[k + 3], s);
      }
      out = h / (1.f + __expf(-s));       // h is binary -> sigmoid(score)*H
    }
    Hatt[(size_t)n * NE + e] = out;
  }
}

// ---------------- t1 = H_att^T @ (dn ⊙ x)  [NE, DD] ----------------
__global__ void k_gather(const float* __restrict__ Hatt, const float* __restrict__ x,
                         const float* __restrict__ dn, float* __restrict__ t1) {
  int wave = threadIdx.x >> 5, lane = threadIdx.x & 31;
  int tile = blockIdx.x * (blockDim.x >> 5) + wave;
  const int ntD = DD / 16;
  int mt = tile / ntD, nt = tile % ntD;
  int half = lane >> 4, lm = lane & 15;
  int e = mt * 16 + lm;
  int d = nt * 16 + lm;

  v8f c = {};
  v2f a, b;
  {
    int ka = half * 2;
    a.x = Hatt[(size_t)ka * NE + e];
    a.y = Hatt[(size_t)(ka + 1) * NE + e];
    b.x = x[(size_t)ka * DD + d] * dn[ka];
    b.y = x[(size_t)(ka + 1) * DD + d] * dn[ka + 1];
  }
  for (int k0 = 0; k0 < NN - 4; k0 += 4) {
    int ka = k0 + 4 + half * 2;
    v2f an, bn;
    an.x = Hatt[(size_t)ka * NE + e];
    an.y = Hatt[(size_t)(ka + 1) * NE + e];
    bn.x = x[(size_t)ka * DD + d] * dn[ka];
    bn.y = x[(size_t)(ka + 1) * DD + d] * dn[ka + 1];
    int pk = k0 + 32; pk = (pk > NN - 4) ? (NN - 4) : pk;
    __builtin_prefetch(Hatt + (size_t)(pk + half * 2) * NE + e, 0, 3);
    c = wmma_f32(a, b, c);
    a = an; b = bn;
  }
  c = wmma_f32(a, b, c);

  int mb = mt * 16 + half * 8;
  #pragma unroll
  for (int r = 0; r < 8; ++r)
    t1[(size_t)(mb + r) * DD + d] = c[r];
}

// ---------------- t2 = H_att @ (de ⊙ t1)  [NN, DD] ----------------
__global__ void k_aggregate(const float* __restrict__ Hatt, const float* __restrict__ t1,
                            const float* __restrict__ de, float* __restrict__ t2) {
  int wave = threadIdx.x >> 5, lane = threadIdx.x & 31;
  int tile = blockIdx.x * (blockDim.x >> 5) + wave;
  const int ntD = DD / 16;
  int mt = tile / ntD, nt = tile % ntD;
  int half = lane >> 4, lm = lane & 15;
  int m = mt * 16 + lm;                // node row
  int d = nt * 16 + lm;
  const float* Arow = Hatt + (size_t)m * NE;

  v8f c = {};
  v2f a, b;
  {
    int ka = half * 2;
    a = *(const v2f*)(Arow + ka);                     // contiguous b64
    b.x = t1[(size_t)ka * DD + d] * de[ka];
    b.y = t1[(size_t)(ka + 1) * DD + d] * de[ka + 1];
  }
  for (int k0 = 0; k0 < NE - 4; k0 += 4) {
    int ka = k0 + 4 + half * 2;
    v2f an = *(const v2f*)(Arow + ka), bn;
    bn.x = t1[(size_t)ka * DD + d] * de[ka];
    bn.y = t1[(size_t)(ka + 1) * DD + d] * de[ka + 1];
    c = wmma_f32(a, b, c);
    a = an; b = bn;
  }
  c = wmma_f32(a, b, c);

  int mb = mt * 16 + half * 8;
  #pragma unroll
  for (int r = 0; r < 8; ++r)
    t2[(size_t)(mb + r) * DD + d] = c[r];
}

// ---------------- out = (dn ⊙ t2) @ Wl^T + bl  [NN, DO] ----------------
__global__ void k_out(const float* __restrict__ t2, const float* __restrict__ dn,
                      const float* __restrict__ Wl, const float* __restrict__ bl,
                      float* __restrict__ out) {
  int wave = threadIdx.x >> 5, lane = threadIdx.x & 31;
  int tile = blockIdx.x * (blockDim.x >> 5) + wave;
  const int ntD = DO / 16;
  int mt = tile / ntD, nt = tile % ntD;
  int half = lane >> 4, lm = lane & 15;
  int m = mt * 16 + lm;
  int j = nt * 16 + lm;
  float sm = dn[m];
  const float* Arow = t2 + (size_t)m * DD;
  const float* Brow = Wl + (size_t)j * DD;
  v8f c = {};
  #pragma unroll
  for (int k0 = 0; k0 < DD; k0 += 4) {
    int ka = k0 + half * 2;
    v2f a = *(const v2f*)(Arow + ka);     // contiguous b64
    v2f b = *(const v2f*)(Brow + ka);
    a.x *= sm; a.y *= sm;
    c = wmma_f32(a, b, c);
  }
  int mb = mt * 16 + half * 8;
  float bj = bl[j];
  #pragma unroll
  for (int r = 0; r < 8; ++r)
    out[(size_t)(mb + r) * DO + j] = c[r] + bj;
}

extern "C" void kernel_launch(void* const* d_in, const int* in_sizes, int n_in,
                              void* d_out, int out_size, void* d_ws, size_t ws_size,
                              hipStream_t stream) {
  const float* x  = (const float*)d_in[0];   // [NN, DD]
  const float* H  = (const float*)d_in[1];   // [NN, NE]
  const float* W1 = (const float*)d_in[2];   // [HH, HH]
  const float* b1 = (const float*)d_in[3];   // [HH]
  const float* W2 = (const float*)d_in[4];   // [1, HH]
  const float* b2 = (const float*)d_in[5];   // [1]
  const float* Wl = (const float*)d_in[6];   // [DO, DD]
  const float* bl = (const float*)d_in[7];   // [DO]
  float* out = (float*)d_out;                // [NN, DO]

  float* ws  = (float*)d_ws;
  float* ef   = ws;  ws += NE * DD;
  float* np   = ws;  ws += NN * HH;
  float* ep   = ws;  ws += NE * HH;
  float* Hatt = ws;  ws += (size_t)NN * NE;
  float* t1   = ws;  ws += NE * DD;
  float* t2   = ws;  ws += NN * DD;
  float* dn   = ws;  ws += NN;
  float* de   = ws;  ws += NE;

  k_deg_n<<<NN / 8, 256, 0, stream>>>(H, dn);
  k_deg_e<<<NE / 256, 256, 0, stream>>>(H, de);
  k_edge_feat<<<64, 128, 0, stream>>>(H, x, ef);          // 256 tiles
  k_proj<<<384, 128, 0, stream>>>(x, ef, W1, np, ep);     // 1536 tiles
  k_score<<<NN, 256, 0, stream>>>(H, np, ep, b1, W2, b2, Hatt);
  k_gather<<<64, 128, 0, stream>>>(Hatt, x, dn, t1);      // 256 tiles
  k_aggregate<<<128, 128, 0, stream>>>(Hatt, t1, de, t2); // 512 tiles
  k_out<<<128, 128, 0, stream>>>(t2, dn, Wl, bl, out);    // 512 tiles
}